// RVSMLLoss_24507083391274
// MI455X (gfx1250) — compile-verified
//
#include <hip/hip_runtime.h>
#include <hip/hip_bf16.h>

// CDNA5 / gfx1250: wave32, WMMA 16x16 shapes.
typedef __attribute__((ext_vector_type(2))) float v2f;
typedef __attribute__((ext_vector_type(8))) float v8f;

#define BLOCK 256
#define WAVES_PER_BLOCK (BLOCK / 32)
#define LGATHER 16
#define LABSTRIDE 17

// Cross-lane (wave32) sum using the matrix pipe.
// A (16x4 f32): lane l<16 holds A[l][0]=x, A[l][1]=0; lane l>=16 holds A[l-16][2]=x, A[l-16][3]=0.
// B = all-ones (layout independent). D[m][n] = x_m + x_{m+16} (replicated over n).
// Each lane sums its 8 D VGPRs (= sum over half the rows), then xor-16 shuffle
// combines halves -> every lane holds the full 32-lane sum.
__device__ inline float wave_reduce_sum(float x) {
#if __has_builtin(__builtin_amdgcn_wmma_f32_16x16x4_f32)
  v2f a; a[0] = x;    a[1] = 0.0f;
  v2f b; b[0] = 1.0f; b[1] = 1.0f;
  v8f c = {};
  v8f d = __builtin_amdgcn_wmma_f32_16x16x4_f32(
      /*neg_a=*/false, a, /*neg_b=*/false, b,
      /*c_mod=*/(short)0, c, /*reuse_a=*/false, /*reuse_b=*/false);
  float s = d[0] + d[1] + d[2] + d[3] + d[4] + d[5] + d[6] + d[7];
  s += __shfl_xor(s, 16, 32);
  return s;
#else
  for (int off = 16; off > 0; off >>= 1) x += __shfl_xor(x, off, 32);
  return x;
#endif
}

// One block per row: streaming sum-of-squares over D floats (float4 lanes,
// contiguous per wave), then combine with weights + gathered values.
// row_out[row] = (sq + 1) * sum_j w_j  -  2 * sum_j w_j * g_j
__global__ void __launch_bounds__(BLOCK)
row_partial_kernel(const float* __restrict__ inputs,
                   const float* __restrict__ labels,
                   float* __restrict__ row_out,
                   int D) {
  const int row = blockIdx.x;
  const int d4 = D >> 2;  // D is a multiple of 4 (16000)
  const float4* __restrict__ rp =
      reinterpret_cast<const float4*>(inputs + (size_t)row * (size_t)D);

  float4 acc4 = make_float4(0.f, 0.f, 0.f, 0.f);
  for (int i = threadIdx.x; i < d4; i += BLOCK) {
    // Pull the next block-stride chunk toward the caches (global_prefetch_b8);
    // speculative: OOB translations are silently dropped.
    __builtin_prefetch(rp + i + BLOCK, 0, 0);
    float4 v = rp[i];  // global_load_b128
    acc4.x = fmaf(v.x, v.x, acc4.x);
    acc4.y = fmaf(v.y, v.y, acc4.y);
    acc4.z = fmaf(v.z, v.z, acc4.z);
    acc4.w = fmaf(v.w, v.w, acc4.w);
  }
  float acc = (acc4.x + acc4.y) + (acc4.z + acc4.w);

  // Divergence-free point: EXEC all ones, safe for WMMA.
  float wsum = wave_reduce_sum(acc);

  __shared__ float lds_part[WAVES_PER_BLOCK];
  if ((threadIdx.x & 31) == 0) lds_part[threadIdx.x >> 5] = wsum;
  __syncthreads();

  if (threadIdx.x == 0) {
    float sq = 0.f;
    #pragma unroll
    for (int w = 0; w < WAVES_PER_BLOCK; ++w) sq += lds_part[w];

    const float* __restrict__ lab = labels + (size_t)row * LABSTRIDE;
    const int cls = (int)lab[LGATHER];
    const float* __restrict__ g =
        inputs + (size_t)row * (size_t)D + (size_t)cls * LGATHER;

    float wacc = 0.f, wg = 0.f;
    #pragma unroll
    for (int j = 0; j < LGATHER; ++j) {
      const float wj = lab[j];
      wacc += wj;
      wg = fmaf(wj, g[j], wg);
    }
    row_out[row] = (sq + 1.0f) * wacc - 2.0f * wg;
  }
}

// Deterministic final reduce of the per-row partials; out = sum / n.
__global__ void __launch_bounds__(BLOCK)
final_reduce_kernel(const float* __restrict__ part, float* __restrict__ out, int n) {
  float acc = 0.f;
  for (int i = threadIdx.x; i < n; i += BLOCK) acc += part[i];

  float s = wave_reduce_sum(acc);

  __shared__ float lds_part[WAVES_PER_BLOCK];
  if ((threadIdx.x & 31) == 0) lds_part[threadIdx.x >> 5] = s;
  __syncthreads();

  if (threadIdx.x == 0) {
    float t = 0.f;
    #pragma unroll
    for (int w = 0; w < WAVES_PER_BLOCK; ++w) t += lds_part[w];
    out[0] = t / (float)n;
  }
}

extern "C" void kernel_launch(void* const* d_in, const int* in_sizes, int n_in,
                              void* d_out, int out_size, void* d_ws, size_t ws_size,
                              hipStream_t stream) {
  const float* inputs = (const float*)d_in[0];  // (N, D) f32
  const float* labels = (const float*)d_in[1];  // (N, 17) f32: 16 weights + cls
  float* out = (float*)d_out;                   // (1,) f32
  float* partial = (float*)d_ws;                // N f32 scratch

  const int N = in_sizes[1] / LABSTRIDE;        // 4096
  const int D = in_sizes[0] / N;                // 16000

  row_partial_kernel<<<N, BLOCK, 0, stream>>>(inputs, labels, partial, D);
  final_reduce_kernel<<<1, BLOCK, 0, stream>>>(partial, out, N);
}